// SimpleTOFPredictor_20340965114496
// MI455X (gfx1250) — compile-verified
//
#include <hip/hip_runtime.h>
#include <hip/hip_bf16.h>
#include <math.h>

// Problem constants (match reference)
#define B1C 4
#define B2C 1024
#define RC  64
#define TC  8192
#define LC  256

typedef __attribute__((ext_vector_type(2))) float v2f;
typedef __attribute__((ext_vector_type(8))) float v8f;

// Exact f32 wave32 sum using V_WMMA_F32_16X16X4_F32.
// A-matrix (16x4, documented layout): VGPR0 holds K=0 (lanes 0-15) / K=2 (lanes 16-31),
// VGPR1 holds K=1 / K=3.  With A = (v, 0) and B = all-ones (layout-independent),
// D[m][n] = v_m + v_{m+16}.  D layout: VGPR j holds M=j (lanes 0-15) / M=j+8 (lanes 16-31),
// so per-lane sum of the 8 D VGPRs gives the half-wave sums; one xor-16 shuffle finishes.
__device__ __forceinline__ float wave_sum_f32(float v) {
    v2f a; a.x = v;    a.y = 0.0f;
    v2f b; b.x = 1.0f; b.y = 1.0f;
    v8f c = {};
    v8f d = __builtin_amdgcn_wmma_f32_16x16x4_f32(
        /*neg_a=*/false, a, /*neg_b=*/false, b,
        /*c_mod=*/(short)0, c, /*reuse_a=*/false, /*reuse_b=*/false);
    float p = ((d[0] + d[1]) + (d[2] + d[3])) + ((d[4] + d[5]) + (d[6] + d[7]));
    p += __shfl_xor(p, 16, 32);
    return p;
}

__global__ __launch_bounds__(256)
void tof_predictor_kernel(const float* __restrict__ rec,    // (B1,R,T)
                          const float* __restrict__ samp,   // (B1,B2,3)
                          const float* __restrict__ emit,   // (3,)
                          const float* __restrict__ rcv,    // (R,3)
                          float* __restrict__ out) {        // (B1,B2)
    const int b   = blockIdx.x;          // flattened (b1,b2), 0..4095
    const int b1  = b >> 10;             // / B2C
    const int tid = threadIdx.x;         // == l (crop sample index)

    __shared__ float s_start[RC];
    __shared__ float s_redA[8];
    __shared__ float s_redB[8];

    // --- per-block: 64 fractional start indices into LDS -------------------
    const float sx = samp[(size_t)b * 3 + 0];
    const float sy = samp[(size_t)b * 3 + 1];
    const float sz = samp[(size_t)b * 3 + 2];
    if (tid < RC) {
        const float ex = emit[0], ey = emit[1], ez = emit[2];
        float dx = sx - ex, dy = sy - ey, dz = sz - ez;
        const float de = sqrtf(dx * dx + dy * dy + dz * dz);
        const float rx = rcv[tid * 3 + 0];
        const float ry = rcv[tid * 3 + 1];
        const float rz = rcv[tid * 3 + 2];
        dx = sx - rx; dy = sy - ry; dz = sz - rz;
        const float dr = sqrtf(dx * dx + dy * dy + dz * dz);
        s_start[tid] = (de + dr) * (96000.0f / 343.0f);   // FS / C
    }
    __syncthreads();

    // Hann half-window: 0.5 - 0.5*cos(linspace(0,pi,L)), step = pi/(L-1)
    const float w = 0.5f - 0.5f * cosf(3.14159265358979323846f * (float)tid * (1.0f / 255.0f));

    // --- gather + interpolate, accumulate per-l sums over receivers --------
    // Lanes map to consecutive l, so i0 = floor(start_r) + l: loads are coalesced,
    // frac is wave-uniform per receiver.  All 64 windows stream through L2.
    float s1 = 0.0f, s2 = 0.0f;
    const float lf = (float)tid;
    const float* __restrict__ recb = rec + (size_t)b1 * (RC * TC);
    #pragma unroll 4
    for (int r = 0; r < RC; ++r) {
        const float idxf = s_start[r] + lf;
        int i0 = (int)floorf(idxf);
        i0 = min(max(i0, 0), TC - 2);
        const float frac = idxf - (float)i0;
        const float* __restrict__ p = recb + (size_t)r * TC;
        const float v0 = p[i0];
        const float v1 = p[i0 + 1];
        const float v  = fmaf(frac, v1 - v0, v0);
        const float xw = v * w;
        s1 += xw;
        s2 = fmaf(xw, xw, s2);
    }

    // Per-thread (= per-l) contributions:
    //   num contrib: s2_l                (-> mean over R*L)
    //   den contrib: s2_l - s1_l^2 / R   (-> /(R-1) per l, mean over L)
    const float aval = s2;
    const float bval = fmaf(-s1 * (1.0f / 64.0f), s1, s2);

    // --- wave reduction via WMMA (uniform flow, EXEC all-ones) -------------
    const float asum = wave_sum_f32(aval);
    const float bsum = wave_sum_f32(bval);

    const int wid = tid >> 5;
    if ((tid & 31) == 0) { s_redA[wid] = asum; s_redB[wid] = bsum; }
    __syncthreads();

    if (tid == 0) {
        float sa = 0.0f, sb = 0.0f;
        #pragma unroll
        for (int i = 0; i < 8; ++i) { sa += s_redA[i]; sb += s_redB[i]; }
        const float num = sa * (1.0f / (64.0f * 256.0f));          // / (R*L)
        const float den = sb * (1.0f / (63.0f * 256.0f)) + 0.001f; // / ((R-1)*L) + eps
        out[b] = num / den;
    }
}

extern "C" void kernel_launch(void* const* d_in, const int* in_sizes, int n_in,
                              void* d_out, int out_size, void* d_ws, size_t ws_size,
                              hipStream_t stream) {
    (void)in_sizes; (void)n_in; (void)out_size; (void)d_ws; (void)ws_size;
    const float* rec  = (const float*)d_in[0];  // recordings        (4,64,8192)
    const float* samp = (const float*)d_in[1];  // sample_locations  (4,1024,3)
    const float* emit = (const float*)d_in[2];  // emitter_location  (3,)
    const float* rcv  = (const float*)d_in[3];  // receiver_locations(64,3)
    float* out = (float*)d_out;                 // (4,1024) f32

    tof_predictor_kernel<<<dim3(B1C * B2C), dim3(256), 0, stream>>>(rec, samp, emit, rcv, out);
}